// MultiNibbleMulStage_62380105007216
// MI455X (gfx1250) — compile-verified
//
#include <hip/hip_runtime.h>
#include <hip/hip_bf16.h>
#include <stdint.h>

// Closed-form of the 4-stage pipeline (all weights are baked constants):
//   t_f   = t - op*t + bJ          (TEMP,   ch 3)   bJ = x[b, 3, NIB_B=1]
//   r_f   = r + a*t_f              (RAW_SUM,ch 4)   a  = x[b, n, NIB_A=0]
//   res_f = res + (n>=3 ? r_f[b,n-3] : 0)  (RESULT, ch 5)
// every other channel of the (B=131072, N=8, D=64) tensor is copied through.
// => pure HBM stream (512 MB total, ~22 us floor at 23.3 TB/s). Use the CDNA5
// async global<->LDS path with non-temporal cache hints (single-pass stream).

namespace {

constexpr unsigned kDim = 64;           // floats per position
constexpr unsigned kBytesPerPos = 256;  // kDim * 4
constexpr unsigned kPosPerBlock = 256;  // one thread per (b, n) position

__global__ __launch_bounds__(kPosPerBlock) void nibble_pipeline_kernel(
    const float* __restrict__ x, float* __restrict__ out) {
  // 64 KB staging tile: 256 positions * 64 floats
  __shared__ float tile[kPosPerBlock * kDim];

  const unsigned tid = threadIdx.x;                       // position within block
  const unsigned pos = blockIdx.x * kPosPerBlock + tid;   // global (b*8+n)
  const unsigned gbyte = pos * kBytesPerPos;              // < 2^28, fits i32
  // Low 32 bits of a flat shared pointer == LDS byte address (aperture rule);
  // verified in disasm: lowers to exactly tid*256.
  const unsigned ldsb = (unsigned)(uintptr_t)(&tile[tid * kDim]);

  // ---- async stream HBM -> LDS (ASYNCcnt path), 16 B/lane per op, NT ----
#pragma unroll
  for (int c = 0; c < 16; ++c) {
    asm volatile("global_load_async_to_lds_b128 %0, %1, %2 offset:%3 th:TH_LOAD_NT"
                 :
                 : "v"(ldsb), "v"(gbyte), "s"(x), "i"(c * 16)
                 : "memory");
  }
  asm volatile("s_wait_asynccnt 0" ::: "memory");

  // ---- scalar fix-ups (all cross-position refs are intra-wave: 32 lanes
  //      cover exactly 4 whole batch elements, N=8; no barrier needed) ----
  const unsigned ti = tid * kDim;
  const unsigned n = tid & 7u;

  const float a   = tile[ti + 0];   // NIB_A
  const float t   = tile[ti + 3];   // TEMP
  const float r   = tile[ti + 4];   // RAW_SUM
  const float res = tile[ti + 5];   // RESULT
  const float op  = tile[ti + 10];  // OP_START + MUL

  const unsigned tj = ((tid & ~7u) | 3u) * kDim;  // row J=3 of same batch
  const float bJ = tile[tj + 1];                  // NIB_B at J (never modified)

  const bool has_prev = (n >= 3u);
  const unsigned tp_base = has_prev ? (tid - 3u) * kDim : ti;  // safe index
  const float ap  = tile[tp_base + 0];
  const float tp  = tile[tp_base + 3];
  const float rp  = tile[tp_base + 4];
  const float opp = tile[tp_base + 10];

  const float t_f   = (t - op * t) + bJ;
  const float r_f   = r + a * t_f;
  const float tp_f  = (tp - opp * tp) + bJ;   // same bJ: n-3 is same batch
  const float rp_f  = rp + ap * tp_f;
  const float res_f = has_prev ? (res + rp_f) : res;

  // All ds_loads above precede these ds_stores in the single wave-lockstep
  // instruction stream, so no RAW hazard across lanes.
  tile[ti + 3] = t_f;
  tile[ti + 4] = r_f;
  tile[ti + 5] = res_f;

  // ---- async stream LDS -> HBM, NT ----
  asm volatile("s_wait_dscnt 0" ::: "memory");  // patches committed to LDS
#pragma unroll
  for (int c = 0; c < 16; ++c) {
    asm volatile("global_store_async_from_lds_b128 %1, %0, %2 offset:%3 th:TH_STORE_NT"
                 :
                 : "v"(ldsb), "v"(gbyte), "s"(out), "i"(c * 16)
                 : "memory");
  }
  asm volatile("s_wait_asynccnt 0" ::: "memory");
}

}  // namespace

extern "C" void kernel_launch(void* const* d_in, const int* in_sizes, int n_in,
                              void* d_out, int out_size, void* d_ws, size_t ws_size,
                              hipStream_t stream) {
  (void)n_in; (void)d_ws; (void)ws_size; (void)out_size;
  const float* x = (const float*)d_in[0];  // (B, N, D) float32; weights are baked
  float* out = (float*)d_out;

  const unsigned total = (unsigned)in_sizes[0];          // B*N*D
  const unsigned positions = total / kDim;               // B*N = 1,048,576
  const unsigned blocks = positions / kPosPerBlock;      // 4096 (exact cover)

  nibble_pipeline_kernel<<<blocks, kPosPerBlock, 0, stream>>>(x, out);
}